// FFM_56813827391600
// MI455X (gfx1250) — compile-verified
//
#include <hip/hip_runtime.h>

#define FNUM 8
#define VOCAB 100000
#define DIM 16
#define SEQL 50

typedef __attribute__((ext_vector_type(2))) float v2f;
typedef __attribute__((ext_vector_type(8))) float v8f;

#if defined(__has_builtin)
#  if __has_builtin(__builtin_amdgcn_wmma_f32_16x16x4_f32)
#    define USE_WMMA 1
#  endif
#endif
#ifndef USE_WMMA
#  define USE_WMMA 0
#endif

// Upper-triangle pairs (i<j), i in high nibble, j in low nibble. 28 pairs.
__constant__ unsigned char PAIR_IJ[28] = {
    0x01,0x02,0x03,0x04,0x05,0x06,0x07,
    0x12,0x13,0x14,0x15,0x16,0x17,
    0x23,0x24,0x25,0x26,0x27,
    0x34,0x35,0x36,0x37,
    0x45,0x46,0x47,
    0x56,0x57,
    0x67};

__global__ __launch_bounds__(256) void ffm_kernel(
    const int* __restrict__ x_seq,        // [B, 50]
    const int* __restrict__ x_cat,        // [B, 7]
    const float* __restrict__ ffm_tables, // [8, 8, V, 16]
    const float* __restrict__ lin_tables, // [8, V]
    const float* __restrict__ dense_w,    // [8]
    const float* __restrict__ dense_b,    // [1]
    float* __restrict__ out)              // [B]
{
    const int b   = blockIdx.x;
    const int tid = threadIdx.x;

    __shared__ int s_idx[SEQL];
    __shared__ int s_cat[FNUM - 1];
    __shared__ __align__(16) float sP[FNUM * FNUM * DIM]; // P[i][j][d], 4 KB
    __shared__ __align__(16) float sZero[DIM];
    __shared__ float sRed[SEQL];
    __shared__ float sZ[FNUM];

    // ---- stage indices into LDS ----
    if (tid < SEQL)                        s_idx[tid]      = x_seq[b * SEQL + tid];
    if (tid >= 64 && tid < 64 + FNUM - 1)  s_cat[tid - 64] = x_cat[b * (FNUM - 1) + (tid - 64)];
    if (tid >= 96 && tid < 96 + DIM)       sZero[tid - 96] = 0.0f;
    __syncthreads();

    // ---- Phase A: sequence field, mean-pool under all 8 target tables ----
    // wave w handles table j=w; lane = lsub*4 + d4; each lane loads one float4
    // (16B) of a 64B row -> full dwordx4 gathers, 8 rows per wave per step.
    {
        const int j    = tid >> 5;   // target table 0..7
        const int lane = tid & 31;
        const int lsub = lane >> 2;  // 0..7: which sequence slot in this step
        const int d4   = lane & 3;   // which float4 chunk of the row
        const float* tab0 = ffm_tables + (size_t)j * VOCAB * DIM; // i=0 block
        float ax = 0.f, ay = 0.f, az = 0.f, aw = 0.f;
        for (int l = lsub; l < SEQL; l += 8) {
            const int idx = s_idx[l];
            const float4 r = ((const float4*)(tab0 + (size_t)idx * DIM))[d4];
            ax += r.x; ay += r.y; az += r.z; aw += r.w;
        }
        // reduce across the 8 lsub groups (stride-4 lanes) within the wave
        for (int off = 16; off >= 4; off >>= 1) {
            ax += __shfl_xor(ax, off, 32);
            ay += __shfl_xor(ay, off, 32);
            az += __shfl_xor(az, off, 32);
            aw += __shfl_xor(aw, off, 32);
        }
        if (lsub == 0) {
            const float inv = 1.0f / (float)SEQL;
            ((float4*)&sP[j * DIM])[d4] =
                make_float4(ax * inv, ay * inv, az * inv, aw * inv);
        }
    }

    // ---- Phase B: categorical fields, 7x8 rows of 64B each ----
    if (tid < 224) {
        const int f  = 1 + (tid >> 5);   // field 1..7
        const int j  = (tid >> 2) & 7;   // target table
        const int d4 = tid & 3;
        const int idx = s_cat[f - 1];
        const float4 r =
            ((const float4*)(ffm_tables + ((size_t)(f * FNUM + j) * VOCAB + idx) * DIM))[d4];
        ((float4*)&sP[(f * FNUM + j) * DIM])[d4] = r;
    }

    // ---- Phase C: linear-term gathers ----
    if (tid < SEQL) sRed[tid] = lin_tables[s_idx[tid]];
    if (tid >= 64 && tid < 64 + FNUM - 1) {
        const int f = tid - 63;  // 1..7
        sZ[f] = lin_tables[(size_t)f * VOCAB + s_cat[f - 1]];
    }
    __syncthreads();

    // ---- Phase D: pairwise interactions + final scalar (wave 0 only) ----
    if (tid < 32) {
        const int lane = tid;
        const int m    = lane & 15;        // pair row within 16-group
        const int kh   = (lane >> 4) << 1; // K sub-offset per ISA f32 layout
        float ffm;
#if USE_WMMA
        // D = A x B^T accumulated over K=16 (4 chained k=4 WMMAs) for two
        // groups of 16 pairs; diag(D) are the 28 pair dot products.
        v8f c = {0.f, 0.f, 0.f, 0.f, 0.f, 0.f, 0.f, 0.f};
#pragma unroll
        for (int g = 0; g < 2; ++g) {
            const int q = g * 16 + m;
            const float* rowA = sZero;
            const float* rowB = sZero;
            if (q < 28) {
                const int pij = PAIR_IJ[q];
                const int i = pij >> 4, j = pij & 15;
                rowA = &sP[(i * FNUM + j) * DIM];
                rowB = &sP[(j * FNUM + i) * DIM];
            }
#pragma unroll
            for (int k0 = 0; k0 < DIM; k0 += 4) {
                v2f a, bb;
                a[0]  = rowA[k0 + kh];
                a[1]  = rowA[k0 + kh + 1];
                bb[0] = rowB[k0 + kh];
                bb[1] = rowB[k0 + kh + 1];
                c = __builtin_amdgcn_wmma_f32_16x16x4_f32(
                        false, a, false, bb, (short)0, c, false, false);
            }
        }
        // extract diagonal: p<8 -> c[p] at lane p; p>=8 -> c[p-8] at lane p+16
        const int sel = (lane < 8) ? lane : ((lane >= 24) ? (lane - 24) : -1);
        float t = 0.f;
#pragma unroll
        for (int r = 0; r < 8; ++r) t += (sel == r) ? c[r] : 0.f;
        for (int off = 16; off; off >>= 1) t += __shfl_xor(t, off, 32);
        ffm = t;
#else
        // scalar fallback: lane p computes pair p's dot
        float t = 0.f;
        if (lane < 28) {
            const int pij = PAIR_IJ[lane];
            const int i = pij >> 4, j = pij & 15;
            const float* rowA = &sP[(i * FNUM + j) * DIM];
            const float* rowB = &sP[(j * FNUM + i) * DIM];
#pragma unroll
            for (int d = 0; d < DIM; ++d) t += rowA[d] * rowB[d];
        }
        for (int off = 16; off; off >>= 1) t += __shfl_xor(t, off, 32);
        ffm = t;
#endif
        if (lane == 0) {
            float z0 = 0.f;
            for (int l = 0; l < SEQL; ++l) z0 += sRed[l];
            z0 *= (1.0f / (float)SEQL);
            float acc = dense_b[0] + z0 * dense_w[0];
#pragma unroll
            for (int f = 1; f < FNUM; ++f) acc += sZ[f] * dense_w[f];
            const float lin = acc > 0.f ? acc : 0.f;
            const float x = lin + ffm;
            out[b] = 1.0f / (1.0f + __expf(-x));
        }
    }
}

extern "C" void kernel_launch(void* const* d_in, const int* in_sizes, int n_in,
                              void* d_out, int out_size, void* d_ws, size_t ws_size,
                              hipStream_t stream) {
    const int*   x_seq      = (const int*)d_in[0];
    const int*   x_cat      = (const int*)d_in[1];
    const float* ffm_tables = (const float*)d_in[2];
    const float* lin_tables = (const float*)d_in[3];
    const float* dense_w    = (const float*)d_in[4];
    const float* dense_b    = (const float*)d_in[5];
    float*       out        = (float*)d_out;

    const int B = in_sizes[0] / SEQL;   // 4096
    ffm_kernel<<<B, 256, 0, stream>>>(x_seq, x_cat, ffm_tables, lin_tables,
                                      dense_w, dense_b, out);
}